// GCN_32332513804701
// MI455X (gfx1250) — compile-verified
//
#include <hip/hip_runtime.h>

typedef __attribute__((ext_vector_type(2))) float v2f;
typedef __attribute__((ext_vector_type(8))) float v8f;

#define FDIM  128
#define NCLS  40
#define NCLSP 48   // 40 padded to 3 tiles of 16

// ---------------- degree / symmetric norm ----------------

__global__ void deg_init_k(float* deg, int n) {
  int i = blockIdx.x * blockDim.x + threadIdx.x;
  if (i < n) deg[i] = 1.0f;  // self-loop contribution
}

__global__ void deg_count_k(const int* __restrict__ dst, float* deg, int nE) {
  int e = blockIdx.x * blockDim.x + threadIdx.x;
  if (e < nE) atomicAdd(&deg[dst[e]], 1.0f);
}

__global__ void dinv_k(float* deg, int n) {  // in place: deg -> rsqrt(deg)
  int i = blockIdx.x * blockDim.x + threadIdx.x;
  if (i < n) deg[i] = rsqrtf(deg[i]);        // deg >= 1 always (self-loops)
}

// ---------------- dense GEMM: Y[nrows,128] = (relu?)X[nrows,128] @ W[128,128] ----------------
// 8 waves/block, each wave owns a 16-row strip, 8 N-tiles of 16x16 f32 WMMA.

template <bool RELU_IN>
__global__ __launch_bounds__(256) void gemm128_k(const float* __restrict__ X,
                                                 const float* __restrict__ W,
                                                 float* __restrict__ Y, int nrows) {
  __shared__ float Ws[FDIM * FDIM];  // 64 KB
  const int tid = threadIdx.x;
  for (int i = tid * 4; i < FDIM * FDIM; i += 256 * 4)
    *(float4*)&Ws[i] = *(const float4*)&W[i];
  __syncthreads();

  const int wave = tid >> 5;
  const int lane = tid & 31;
  const int row0 = (blockIdx.x * 8 + wave) * 16;
  if (row0 >= nrows) return;  // wave-uniform: nrows % 16 == 0, EXEC stays all-ones

  const int m  = lane & 15;          // M (A) / N (B,C) index within tile
  const int kh = (lane >> 4) << 1;   // 0 or 2: K sub-pair per half-wave
  const float* __restrict__ arow = X + (size_t)(row0 + m) * FDIM;

  v8f acc[8] = {};
  for (int k = 0; k < FDIM; k += 4) {
    v2f a = *(const v2f*)(arow + k + kh);
    if (RELU_IN) { a[0] = fmaxf(a[0], 0.0f); a[1] = fmaxf(a[1], 0.0f); }
    const float* brow0 = &Ws[(k + kh) * FDIM];
    const float* brow1 = &Ws[(k + kh + 1) * FDIM];
#pragma unroll
    for (int t = 0; t < 8; ++t) {
      const int col = t * 16 + m;
      v2f b;
      b[0] = brow0[col];
      b[1] = brow1[col];
      acc[t] = __builtin_amdgcn_wmma_f32_16x16x4_f32(false, a, false, b,
                                                     (short)0, acc[t], false, false);
    }
  }

  const int rbase = row0 + ((lane >> 4) << 3);  // C: lanes 16-31 hold M+8
#pragma unroll
  for (int t = 0; t < 8; ++t) {
    const int col = t * 16 + m;
#pragma unroll
    for (int i = 0; i < 8; ++i)
      Y[(size_t)(rbase + i) * FDIM + col] = acc[t][i];
  }
}

// ---------------- classifier GEMM: Y[nrows,40] = relu(A) @ Wl[128,40] + bl ----------------

__global__ __launch_bounds__(256) void gemm_cls_k(const float* __restrict__ A,
                                                  const float* __restrict__ W,
                                                  const float* __restrict__ bias,
                                                  float* __restrict__ Y, int nrows) {
  __shared__ float Ws[FDIM * NCLSP];  // zero-padded to 48 cols
  for (int i = threadIdx.x; i < FDIM * NCLSP; i += 256) {
    int k = i / NCLSP, c = i - k * NCLSP;
    Ws[i] = (c < NCLS) ? W[k * NCLS + c] : 0.0f;
  }
  __syncthreads();

  const int tid = threadIdx.x;
  const int wave = tid >> 5;
  const int lane = tid & 31;
  const int row0 = (blockIdx.x * 8 + wave) * 16;
  if (row0 >= nrows) return;

  const int m  = lane & 15;
  const int kh = (lane >> 4) << 1;
  const float* __restrict__ arow = A + (size_t)(row0 + m) * FDIM;

  v8f acc[3] = {};
  for (int k = 0; k < FDIM; k += 4) {
    v2f a = *(const v2f*)(arow + k + kh);
    a[0] = fmaxf(a[0], 0.0f);
    a[1] = fmaxf(a[1], 0.0f);
#pragma unroll
    for (int t = 0; t < 3; ++t) {
      const int col = t * 16 + m;
      v2f b;
      b[0] = Ws[(k + kh) * NCLSP + col];
      b[1] = Ws[(k + kh + 1) * NCLSP + col];
      acc[t] = __builtin_amdgcn_wmma_f32_16x16x4_f32(false, a, false, b,
                                                     (short)0, acc[t], false, false);
    }
  }

  const int rbase = row0 + ((lane >> 4) << 3);
#pragma unroll
  for (int t = 0; t < 3; ++t) {
    const int col = t * 16 + m;
    if (col < NCLS) {
      const float bv = bias[col];
#pragma unroll
      for (int i = 0; i < 8; ++i)
        Y[(size_t)(rbase + i) * NCLS + col] = acc[t][i] + bv;
    }
  }
}

// ---------------- aggregation: out = b + xw*dinv^2 (self loop), then edge scatter ----------------
// one wave per node/edge; 32 lanes x float4 = 128 contiguous floats (512 B)

__global__ void agg_init_k(const float* __restrict__ xw, const float* __restrict__ dinv,
                           const float* __restrict__ bias, float* __restrict__ out, int n) {
  int t = blockIdx.x * blockDim.x + threadIdx.x;
  int node = t >> 5, lane = t & 31;
  if (node >= n) return;
  float s = dinv[node];
  s *= s;
  float4 x = ((const float4*)(xw + (size_t)node * FDIM))[lane];
  float4 b = ((const float4*)bias)[lane];
  float4 o;
  o.x = fmaf(x.x, s, b.x);
  o.y = fmaf(x.y, s, b.y);
  o.z = fmaf(x.z, s, b.z);
  o.w = fmaf(x.w, s, b.w);
  ((float4*)(out + (size_t)node * FDIM))[lane] = o;
}

__global__ void agg_edge_k(const float* __restrict__ xw, const int* __restrict__ src,
                           const int* __restrict__ dst, const float* __restrict__ dinv,
                           float* __restrict__ out, int nE) {
  int t = blockIdx.x * blockDim.x + threadIdx.x;
  int e = t >> 5, lane = t & 31;
  if (e >= nE) return;
  int s = src[e], d = dst[e];
  float w = dinv[s] * dinv[d];
  float4 x = ((const float4*)(xw + (size_t)s * FDIM))[lane];
  float* op = out + (size_t)d * FDIM + lane * 4;
  atomicAdd(op + 0, x.x * w);
  atomicAdd(op + 1, x.y * w);
  atomicAdd(op + 2, x.z * w);
  atomicAdd(op + 3, x.w * w);
}

// ---------------- launch ----------------

extern "C" void kernel_launch(void* const* d_in, const int* in_sizes, int n_in,
                              void* d_out, int out_size, void* d_ws, size_t ws_size,
                              hipStream_t stream) {
  (void)n_in; (void)out_size; (void)ws_size;
  const float* X  = (const float*)d_in[0];
  const int*   EI = (const int*)d_in[1];
  const float* W1 = (const float*)d_in[2];
  const float* b1 = (const float*)d_in[3];
  const float* W2 = (const float*)d_in[4];
  const float* b2 = (const float*)d_in[5];
  const float* Wl = (const float*)d_in[6];
  const float* bl = (const float*)d_in[7];

  const int N = in_sizes[0] / FDIM;
  const int E = in_sizes[1] / 2;
  const int* src = EI;       // edge_indices[0]
  const int* dst = EI + E;   // edge_indices[1]

  float* ws   = (float*)d_ws;
  float* dinv = ws;                                 // N floats
  float* buf1 = ws + N;                             // N*128 floats
  float* buf2 = buf1 + (size_t)N * FDIM;            // N*128 floats

  const int T = 256;
  const int gemm_blocks = (N + 127) / 128;
  const int nodeGrid = (int)(((long long)N * 32 + T - 1) / T);
  const int edgeGrid = (int)(((long long)E * 32 + T - 1) / T);

  // symmetric norm (self-loops folded in)
  deg_init_k<<<(N + T - 1) / T, T, 0, stream>>>(dinv, N);
  deg_count_k<<<(E + T - 1) / T, T, 0, stream>>>(dst, dinv, E);
  dinv_k<<<(N + T - 1) / T, T, 0, stream>>>(dinv, N);

  // layer 1: XW1, then agg (init = bias + self-loop term, edges via f32 atomics)
  gemm128_k<false><<<gemm_blocks, T, 0, stream>>>(X, W1, buf1, N);
  agg_init_k<<<nodeGrid, T, 0, stream>>>(buf1, dinv, b1, buf2, N);
  agg_edge_k<<<edgeGrid, T, 0, stream>>>(buf1, src, dst, dinv, buf2, E);

  // layer 2 (relu fused into GEMM A-load)
  gemm128_k<true><<<gemm_blocks, T, 0, stream>>>(buf2, W2, buf1, N);
  agg_init_k<<<nodeGrid, T, 0, stream>>>(buf1, dinv, b2, buf2, N);
  agg_edge_k<<<edgeGrid, T, 0, stream>>>(buf1, src, dst, dinv, buf2, E);

  // classifier (relu fused, bias in epilogue)
  gemm_cls_k<<<gemm_blocks, T, 0, stream>>>(buf2, Wl, bl, (float*)d_out, N);
}